// VolterraConv2d__67284957659486
// MI455X (gfx1250) — compile-verified
//
#include <hip/hip_runtime.h>

// CDNA5 / gfx1250: fp32 WMMA fragments
typedef float v2f __attribute__((ext_vector_type(2)));
typedef float v8f __attribute__((ext_vector_type(8)));

#define KSS     9      // 3x3 patch
#define NPAIR   45     // combinations_with_replacement(9, 2)
#define KF      56     // per-channel K, padded 54 -> 56 (14 steps of K=4)
#define C_IN    64
#define OC      64
#define HW      64
#define BLOCK_M 128    // pixels per block (two image rows)
#define VSCALE  0.1490711984999860f   // 1/sqrt(45), folded into quadratic features

// live W2 row index (p*9+q for p<=q) for each of the 45 quadratic features
__constant__ int c_pq[NPAIR] = {
     0, 1, 2, 3, 4, 5, 6, 7, 8,
    10,11,12,13,14,15,16,17,
    20,21,22,23,24,25,26,
    30,31,32,33,34,35,
    40,41,42,43,44,
    50,51,52,53,
    60,61,62,
    70,71,
    80
};

__global__ __launch_bounds__(256)
void volterra_wmma_f32_kernel(const float* __restrict__ x,
                              const float* __restrict__ W1,
                              const float* __restrict__ W2,
                              const float* __restrict__ bias_w,
                              float* __restrict__ out)
{
    __shared__ float sA[BLOCK_M][KF + 1];  // 128 x 57 (pad keeps banks clean)
    __shared__ float sB[KF][OC];           // 56 x 64 effective weight panel

    const int tid    = threadIdx.x;
    const int wave   = tid >> 5;       // 0..7
    const int nslice = wave & 3;       // which 16-wide N slice
    const int mgroup = wave >> 2;      // which 64-row half of the M block
    const int lane   = tid & 31;
    const int half   = lane >> 4;      // 0: K={0,1}/rows 0-7 ; 1: K={2,3}/rows 8-15
    const int l      = lane & 15;

    // 128 blocks: image b, two consecutive rows y0, y0+1 (full 64-wide rows)
    const int blk = blockIdx.x;
    const int b   = blk >> 5;
    const int y0  = (blk & 31) << 1;

    v8f acc[4];                         // 4 x (16x16 fp32) accumulator tiles
#pragma unroll
    for (int mt = 0; mt < 4; ++mt) acc[mt] = (v8f){};

    for (int c = 0; c < C_IN; ++c) {
        // ---- stage A panel: 128 pixels x 56 features (threads 0..127) ----
        if (tid < BLOCK_M) {
            const int y  = y0 + (tid >> 6);
            const int xg = tid & 63;
            const float* xc = x + (size_t)(b * C_IN + c) * (HW * HW);
            float v[KSS];
#pragma unroll
            for (int dy = 0; dy < 3; ++dy) {
#pragma unroll
                for (int dx = 0; dx < 3; ++dx) {
                    const int yy = y + dy - 1;
                    const int xx = xg + dx - 1;
                    float t = 0.0f;
                    if (yy >= 0 && yy < HW && xx >= 0 && xx < HW)
                        t = xc[yy * HW + xx];
                    v[dy * 3 + dx] = t;
                }
            }
#pragma unroll
            for (int j = 0; j < KSS; ++j) sA[tid][j] = v[j];
            int idx = KSS;
#pragma unroll
            for (int p = 0; p < KSS; ++p) {
                const float vs = v[p] * VSCALE;   // fold mask-scale into features
#pragma unroll
                for (int q = p; q < KSS; ++q) {
                    sA[tid][idx++] = vs * v[q];
                }
            }
            sA[tid][54] = 0.0f;   // K padding
            sA[tid][55] = 0.0f;
        }

        // ---- stage B panel: pure gather of W1 rows + live W2 rows ----
#pragma unroll
        for (int i = tid; i < KF * OC; i += 256) {   // exactly 14 per thread
            const int r = i >> 6;
            const int n = i & 63;
            float wv = 0.0f;
            if (r < KSS) {
                wv = W1[(c * KSS + r) * OC + n];
            } else if (r < KSS + NPAIR) {
                wv = W2[(c * (KSS * KSS) + c_pq[r - KSS]) * OC + n];
            }
            sB[r][n] = wv;
        }
        __syncthreads();

        // ---- 14 K-steps; each B-frag feeds 4 WMMAs over 4 M-subtiles ----
        const int nbase = nslice * 16 + l;
        const int rbase = mgroup * 64 + l;
#pragma unroll
        for (int kk = 0; kk < KF; kk += 4) {
            const int ka = kk + half * 2;
            v2f bfrag;
            bfrag.x = sB[ka][nbase];
            bfrag.y = sB[ka + 1][nbase];
#pragma unroll
            for (int mt = 0; mt < 4; ++mt) {
                v2f afrag;
                afrag.x = sA[rbase + mt * 16][ka];
                afrag.y = sA[rbase + mt * 16][ka + 1];
                acc[mt] = __builtin_amdgcn_wmma_f32_16x16x4_f32(
                        /*neg_a=*/false, afrag,
                        /*neg_b=*/false, bfrag,
                        /*c_mod=*/(short)0, acc[mt],
                        /*reuse_a=*/false, /*reuse_b=*/false);
            }
        }
        __syncthreads();
    }

    // ---- epilogue: C/D layout — lane l = column N, VGPR e = row (+8 if half) ----
    const int n  = nslice * 16 + l;
    const float bv = bias_w[n];
#pragma unroll
    for (int mt = 0; mt < 4; ++mt) {
#pragma unroll
        for (int e = 0; e < 8; ++e) {
            const int m  = mgroup * 64 + mt * 16 + e + half * 8;  // pixel in block
            const int y  = y0 + (m >> 6);
            const int xx = m & 63;
            out[((size_t)(b * OC + n) * HW + y) * HW + xx] = acc[mt][e] + bv;
        }
    }
}

extern "C" void kernel_launch(void* const* d_in, const int* in_sizes, int n_in,
                              void* d_out, int out_size, void* d_ws, size_t ws_size,
                              hipStream_t stream) {
    (void)in_sizes; (void)n_in; (void)out_size; (void)d_ws; (void)ws_size;
    const float* x  = (const float*)d_in[0];
    const float* W1 = (const float*)d_in[1];
    const float* W2 = (const float*)d_in[2];
    const float* bw = (const float*)d_in[3];
    float* out = (float*)d_out;

    const int B = 4;
    const int blocks = B * (HW / 2);   // 128 blocks, each = 2 image rows
    volterra_wmma_f32_kernel<<<blocks, 256, 0, stream>>>(x, W1, W2, bw, out);
}